// StreamingSelfAttention_36876589204185
// MI455X (gfx1250) — compile-verified
//
#include <hip/hip_runtime.h>
#include <math.h>

// ---------------------------------------------------------------------------
// StreamingSelfAttention for MI455X (gfx1250, wave32, WMMA f32 16x16x4)
//   B=2, S=4096, D=512, H=8, hd=64, LOOKAHEAD=0, LEFT_CONTEXT=128
// Pipeline:  qkv = x @ Wqkv^T + b   (WMMA GEMM, async-LDS double buffered)
//            ctx = banded softmax attention (WMMA, band = 129 keys)
//            out = ctx @ Wout^T + b (same GEMM)
// fp32 end-to-end via V_WMMA_F32_16X16X4_F32 (matches fp32 reference).
// GEMM staging uses GLOBAL_LOAD_ASYNC_TO_LDS_B128 (ASYNCcnt) so the next
// K-tile fill overlaps the current tile's WMMA stream.
// ---------------------------------------------------------------------------

typedef __attribute__((ext_vector_type(2))) float v2f;
typedef __attribute__((ext_vector_type(8))) float v8f;

__device__ __forceinline__ v8f wmma4(v2f a, v2f b, v8f c) {
  // (neg_a, A, neg_b, B, c_mod, C, reuse_a, reuse_b)
  return __builtin_amdgcn_wmma_f32_16x16x4_f32(false, a, false, b, (short)0, c,
                                               false, false);
}

// Async global->LDS copy of 16 bytes per lane. LDS address = low 32 bits of
// the generic shared-memory pointer (ISA 10.2: those bits are the LDS
// byte offset relative to the wave's allocation base).
__device__ __forceinline__ void async_g2l_b128(void* lds, const void* gptr) {
  unsigned       l = (unsigned)(size_t)lds;
  unsigned long long g = (unsigned long long)(size_t)gptr;
  asm volatile("global_load_async_to_lds_b128 %0, %1, off"
               :: "v"(l), "v"(g) : "memory");
}
__device__ __forceinline__ void wait_async_le6() {
  asm volatile("s_wait_asynccnt 0x6" ::: "memory");
}
__device__ __forceinline__ void wait_async_0() {
  asm volatile("s_wait_asynccnt 0x0" ::: "memory");
}

// ---------------------------- constants ------------------------------------
#define SB 2
#define SS 4096
#define SD 512
#define SH 8
#define SHD 64
#define STD 1536   // 3*D, qkv row stride
#define SLC 128    // left context

// ------------------------ generic GEMM: C = A @ W^T + bias ------------------
// A: [M,K] row-major, W: [N,K] row-major, C: [M,N]. M%64==0, N%128==0, K%32==0
#define TM 64
#define TN 128
#define TK 32
#define TKP 36     // padded LDS row stride: 144B (16B-aligned rows, no conflicts)

__global__ __launch_bounds__(256) void gemm_nt_wmma(
    const float* __restrict__ A, const float* __restrict__ W,
    const float* __restrict__ bias, float* __restrict__ C,
    int M, int N, int K) {
  __shared__ float As[2][TM][TKP];
  __shared__ float Ws[2][TN][TKP];

  const int tid = threadIdx.x;
  const int lid = tid & 31;
  const int wv  = tid >> 5;        // wave 0..7
  const int wm  = wv & 1;          // 2 M-strips of 32
  const int wn  = wv >> 1;         // 4 N-strips of 32
  const int m0  = blockIdx.y * TM;
  const int n0  = blockIdx.x * TN;

  const int nloc  = lid & 15;
  const int khalf = (lid >> 4) << 1;   // 0 or 2

  // Per-thread staging coordinates (float4 granules).
  // A tile: 64x32 = 512 granules -> 2/thread; W tile: 128x32 = 1024 -> 4/thread.
  int arow[2], acol[2], wrow[4], wcol[4];
  for (int i = 0; i < 2; ++i) {
    int idx = tid + i * 256;
    arow[i] = idx >> 3;
    acol[i] = (idx & 7) << 2;
  }
  for (int i = 0; i < 4; ++i) {
    int idx = tid + i * 256;
    wrow[i] = idx >> 3;
    wcol[i] = (idx & 7) << 2;
  }

  v8f acc[2][2] = {};

  // Issue 6 async b128 per wave per K-tile: 2 (A) + 4 (W).
  auto stage = [&](int buf, int k0) {
    for (int i = 0; i < 2; ++i)
      async_g2l_b128(&As[buf][arow[i]][acol[i]],
                     &A[(size_t)(m0 + arow[i]) * K + k0 + acol[i]]);
    for (int i = 0; i < 4; ++i)
      async_g2l_b128(&Ws[buf][wrow[i]][wcol[i]],
                     &W[(size_t)(n0 + wrow[i]) * K + k0 + wcol[i]]);
  };

  stage(0, 0);
  const int nkt = K / TK;
  for (int kt = 0; kt < nkt; ++kt) {
    const int cur = kt & 1;
    if (kt + 1 < nkt) {
      stage(cur ^ 1, (kt + 1) * TK);   // prefetch next tile into other buffer
      wait_async_le6();                // retire current tile (in-order), keep 6 in flight
    } else {
      wait_async_0();
    }
    __syncthreads();

    for (int kk = 0; kk < TK; kk += 4) {
      const int kf = kk + khalf;
      v2f af[2], bf[2];
      for (int i = 0; i < 2; ++i)
        af[i] = *(const v2f*)&As[cur][wm * 32 + i * 16 + nloc][kf];
      for (int j = 0; j < 2; ++j)
        bf[j] = *(const v2f*)&Ws[cur][wn * 32 + j * 16 + nloc][kf];
      for (int i = 0; i < 2; ++i)
        for (int j = 0; j < 2; ++j)
          acc[i][j] = wmma4(af[i], bf[j], acc[i][j]);
    }
    __syncthreads();   // all waves done reading buf before it is refilled
  }

  // C/D layout: vgpr r -> M=r (lanes 0-15) / M=r+8 (lanes 16-31), N=lane%16
  const int moff = (lid >> 4) << 3;
  for (int i = 0; i < 2; ++i)
    for (int j = 0; j < 2; ++j) {
      const int n  = n0 + wn * 32 + j * 16 + nloc;
      const float bn = bias[n];
      for (int r = 0; r < 8; ++r) {
        const int m = m0 + wm * 32 + i * 16 + r + moff;
        C[(size_t)m * N + n] = acc[i][j][r] + bn;
      }
    }
}

// -------------------- banded attention (one (b,h,q-tile) per block) --------
// 128 threads = 4 waves; wave w owns query rows [q0+16w, q0+16w+15].
// Keys span [q0-128, q0+63] = 12 tiles of 16. Scores+probs live in LDS.
// qkv (50MB) is L2-resident (192MB L2), so K/V stream straight from global.
__global__ __launch_bounds__(128) void attn_band_wmma(
    const float* __restrict__ qkv, float* __restrict__ ctx) {
  __shared__ float P[64][192 + 4];

  const int tid = threadIdx.x;
  const int lid = tid & 31;
  const int wv  = tid >> 5;                 // 0..3
  const int q0  = blockIdx.x * 64;
  const int h   = blockIdx.y;
  const int bb  = blockIdx.z;

  const size_t rowbase = (size_t)bb * SS;
  const int qoff = h * SHD;
  const int koff = SD + h * SHD;
  const int voff = 2 * SD + h * SHD;

  const int mloc  = lid & 15;
  const int khalf = (lid >> 4) << 1;
  const int moff  = (lid >> 4) << 3;

  // preload Q A-fragments: 16 K-steps of 4 over hd=64
  v2f qa[16];
  {
    const float* qp = qkv + (rowbase + q0 + wv * 16 + mloc) * STD + qoff;
    for (int ks = 0; ks < 16; ++ks)
      qa[ks] = *(const v2f*)&qp[ks * 4 + khalf];
  }

  const float scale = 0.125f;  // 1/sqrt(64)

  // ---- scores: S = (Q K^T) * scale, masked, -> LDS
  for (int t = 0; t < 12; ++t) {
    const int kb = q0 - SLC + t * 16;
    const int keyrow = kb + mloc;                 // this lane's key (B col n)
    const int crow = keyrow < 0 ? 0 : keyrow;     // clamp; masked below
    const float* kp = qkv + (rowbase + crow) * STD + koff;
    v8f acc = {};
    for (int ks = 0; ks < 16; ++ks) {
      v2f bv = *(const v2f*)&kp[ks * 4 + khalf];
      acc = wmma4(qa[ks], bv, acc);
    }
    for (int r = 0; r < 8; ++r) {
      const int m  = r + moff;                    // local query row in tile
      const int qi = q0 + wv * 16 + m;
      const int kj = kb + mloc;                   // key col for this lane
      const bool ok = (kj >= 0) & (kj <= qi) & (kj >= qi - SLC);
      float v = acc[r] * scale;
      P[wv * 16 + m][t * 16 + mloc] = ok ? v : -__builtin_inff();
    }
  }
  __syncthreads();

  // ---- softmax per query row (tid<64: one row each)
  if (tid < 64) {
    float mx = -__builtin_inff();
    for (int j = 0; j < 192; ++j) mx = fmaxf(mx, P[tid][j]);
    float sum = 0.f;
    for (int j = 0; j < 192; ++j) {
      float e = expf(P[tid][j] - mx);
      P[tid][j] = e;
      sum += e;
    }
    const float inv = 1.0f / sum;
    for (int j = 0; j < 192; ++j) P[tid][j] *= inv;
  }
  __syncthreads();

  // ---- ctx = P (64x192) @ V (192x64); wave handles its 16 rows, 4 n-tiles
  v8f acc2[4] = {};
  for (int kk = 0; kk < 192; kk += 4) {
    v2f a = *(const v2f*)&P[wv * 16 + mloc][kk + khalf];
    int kr0 = q0 - SLC + kk + khalf;              // V rows for b.x / b.y
    int r0 = kr0 < 0 ? 0 : kr0;
    int r1 = (kr0 + 1) < 0 ? 0 : (kr0 + 1);
    const float* vp0 = qkv + (rowbase + r0) * STD + voff;
    const float* vp1 = qkv + (rowbase + r1) * STD + voff;
    for (int j = 0; j < 4; ++j) {
      const int n = j * 16 + mloc;
      v2f bv;
      bv.x = vp0[n];
      bv.y = vp1[n];
      acc2[j] = wmma4(a, bv, acc2[j]);
    }
  }

  for (int j = 0; j < 4; ++j)
    for (int r = 0; r < 8; ++r) {
      const int m = q0 + wv * 16 + r + moff;
      const int n = h * SHD + j * 16 + mloc;
      ctx[(rowbase + m) * SD + n] = acc2[j][r];
    }
}

// ---------------------------------------------------------------------------
extern "C" void kernel_launch(void* const* d_in, const int* in_sizes, int n_in,
                              void* d_out, int out_size, void* d_ws,
                              size_t ws_size, hipStream_t stream) {
  (void)in_sizes; (void)n_in; (void)out_size; (void)ws_size;
  const float* x   = (const float*)d_in[0];   // [B,S,D]
  const float* ipw = (const float*)d_in[1];   // [3D,D]
  const float* ipb = (const float*)d_in[2];   // [3D]
  const float* ow  = (const float*)d_in[3];   // [D,D]
  const float* ob  = (const float*)d_in[4];   // [D]
  float* out = (float*)d_out;                 // [B,S,D]

  float* qkv = (float*)d_ws;                              // 8192 x 1536
  float* ctx = qkv + (size_t)SB * SS * STD;                // 8192 x 512

  const int M = SB * SS;  // 8192

  // qkv = x @ Wqkv^T + b      (M=8192, N=1536, K=512)
  gemm_nt_wmma<<<dim3(STD / TN, M / TM), dim3(256), 0, stream>>>(
      x, ipw, ipb, qkv, M, STD, SD);

  // banded attention -> ctx   (grid: S/64 x H x B)
  attn_band_wmma<<<dim3(SS / 64, SH, SB), dim3(128), 0, stream>>>(qkv, ctx);

  // out = ctx @ Wout^T + b    (M=8192, N=512, K=512)
  gemm_nt_wmma<<<dim3(SD / TN, M / TM), dim3(256), 0, stream>>>(
      ctx, ow, ob, out, M, SD, SD);
}